// MultiScalePatternModel_30820685316797
// MI455X (gfx1250) — compile-verified
//
#include <hip/hip_runtime.h>
#include <stdint.h>

// ---------------------------------------------------------------------------
// MultiScalePatternModel, B = 2,000,000, MAX_N = 4
//
// Precompute (in d_ws):
//   PB[s][paddr]  : 3 pattern bits packed in a byte.    sizes 8/64/512/4096 (4680 B)
//   RT[s][k][w]   : position_tables gathered through the fixed bit-permute
//                   addr_{s,k}(w).  w has 3(s+1)+3 bits.
//                   float sizes 5*{64,512,4096,32768} = 187200 floats
// Main kernel: PB + RT[s=0..2] staged in LDS (~96 KB) via async global->LDS;
// RT[s=3] (640 KB) gathered from global (L2-resident).
// ---------------------------------------------------------------------------

#define RT_TOTAL 187200   // floats
#define RT3_BASE 23360    // floats: RT for s=0..2 live in [0, 23360)
#define PB_TOTAL 4680     // bytes
#define PB_W32   1170     // PB_TOTAL / 4

typedef int   vint4_t   __attribute__((ext_vector_type(4)));
typedef float vfloat4_t __attribute__((ext_vector_type(4)));

#if defined(__AMDGCN__) && __has_builtin(__builtin_amdgcn_global_load_async_to_lds_b128) && __has_builtin(__builtin_amdgcn_global_load_async_to_lds_b32)
#define USE_ASYNC_LDS 1
#else
#define USE_ASYNC_LDS 0
#endif

#if USE_ASYNC_LDS
typedef __attribute__((address_space(1))) vint4_t* gptr_b128_t;
typedef __attribute__((address_space(3))) vint4_t* lptr_b128_t;
typedef __attribute__((address_space(1))) int*     gptr_b32_t;
typedef __attribute__((address_space(3))) int*     lptr_b32_t;

__device__ __forceinline__ void g2lds_b128(const void* gp, void* lp) {
  __builtin_amdgcn_global_load_async_to_lds_b128(
      (gptr_b128_t)(uintptr_t)gp,
      (lptr_b128_t)(uint32_t)(uintptr_t)lp, 0, 0);
}
__device__ __forceinline__ void g2lds_b32(const void* gp, void* lp) {
  __builtin_amdgcn_global_load_async_to_lds_b32(
      (gptr_b32_t)(uintptr_t)gp,
      (lptr_b32_t)(uint32_t)(uintptr_t)lp, 0, 0);
}
__device__ __forceinline__ void wait_async_lds() {
#if __has_builtin(__builtin_amdgcn_s_wait_asynccnt)
  __builtin_amdgcn_s_wait_asynccnt(0);
#else
  asm volatile("s_wait_asynccnt 0" ::: "memory");
#endif
}
#endif

// ---------------------------------------------------------------------------
// Setup kernel 1: pack thresholded pattern bits.
// idx space: s0 [0,8), s1 [8,72), s2 [72,584), s3 [584,4680)
// ---------------------------------------------------------------------------
__global__ __launch_bounds__(256) void build_pb_kernel(
    const float* __restrict__ pat, unsigned char* __restrict__ pbg) {
  int idx = blockIdx.x * blockDim.x + threadIdx.x;
  if (idx >= PB_TOTAL) return;
  int s, base;
  if (idx < 8)        { s = 0; base = 0;   }
  else if (idx < 72)  { s = 1; base = 8;   }
  else if (idx < 584) { s = 2; base = 72;  }
  else                { s = 3; base = 584; }
  int paddr = idx - base;
  unsigned bits = 0;
#pragma unroll
  for (int t = 0; t < 3; ++t) {
    float v = pat[(s * 3 + t) * 4096 + paddr];
    bits |= (v > 0.5f ? 1u : 0u) << t;
  }
  pbg[idx] = (unsigned char)bits;
}

// ---------------------------------------------------------------------------
// Setup kernel 2: remapped position tables RT[s][k][w].
// w bit semantics: bit c (c < 3n) == ctx[3n-1-c]  (i.e. w low bits == paddr),
//                  bit 3n+t       == pbits[t].
// idx bases (floats): s0 0, s1 320, s2 2880, s3 23360; wlen 64/512/4096/32768
// ---------------------------------------------------------------------------
__global__ __launch_bounds__(256) void build_rt_kernel(
    const float* __restrict__ pos, const int* __restrict__ conn,
    float* __restrict__ rt) {
  int idx = blockIdx.x * blockDim.x + threadIdx.x;
  if (idx >= RT_TOTAL) return;
  int s, base, wlen;
  if (idx < 320)        { s = 0; base = 0;     wlen = 64;    }
  else if (idx < 2880)  { s = 1; base = 320;   wlen = 512;   }
  else if (idx < 23360) { s = 2; base = 2880;  wlen = 4096;  }
  else                  { s = 3; base = 23360; wlen = 32768; }
  int r = idx - base;
  int k = r / wlen;
  int w = r - k * wlen;
  int in3 = 3 * (s + 1);
  int in_bits = in3 + 3;
  int nb = in_bits < 12 ? in_bits : 12;
  int addr = 0;
  for (int j = 0; j < nb; ++j) {
    int c = conn[(s * 5 + k) * 12 + j] % in_bits;
    int bit = (c < in3) ? ((w >> (in3 - 1 - c)) & 1) : ((w >> c) & 1);
    addr += bit << j;
  }
  rt[idx] = pos[(s * 5 + k) * 4096 + addr];
}

// ---------------------------------------------------------------------------
// Main kernel: grid-stride over items; LDS-staged PB + RT[s<3].
// ---------------------------------------------------------------------------
__global__ __launch_bounds__(1024) void msp_main_kernel(
    const int* __restrict__ tb, const float* __restrict__ rt,
    const unsigned char* __restrict__ pbg, float* __restrict__ out,
    int nitems) {
  __shared__ __align__(16) float rt_lds[RT3_BASE];
  __shared__ unsigned int pb_lds_u32[PB_W32];

  const unsigned int* pbg32 = (const unsigned int*)pbg;
  const float* rt3 = rt + RT3_BASE;

#if USE_ASYNC_LDS
  for (int i = threadIdx.x; i < RT3_BASE / 4; i += blockDim.x)
    g2lds_b128(rt + i * 4, &rt_lds[i * 4]);
  for (int i = threadIdx.x; i < PB_W32; i += blockDim.x)
    g2lds_b32(pbg32 + i, &pb_lds_u32[i]);
  wait_async_lds();
#else
  {
    vfloat4_t* dst = (vfloat4_t*)rt_lds;
    const vfloat4_t* src = (const vfloat4_t*)rt;
    for (int i = threadIdx.x; i < RT3_BASE / 4; i += blockDim.x) dst[i] = src[i];
    for (int i = threadIdx.x; i < PB_W32; i += blockDim.x) pb_lds_u32[i] = pbg32[i];
  }
#endif
  __syncthreads();

  const unsigned char* pb_lds = (const unsigned char*)pb_lds_u32;
  int stride = gridDim.x * blockDim.x;
  for (int i = blockIdx.x * blockDim.x + threadIdx.x; i < nitems; i += stride) {
    const vint4_t* tp = (const vint4_t*)(tb + (size_t)i * 12);
    vint4_t a = __builtin_nontemporal_load(tp + 0);
    vint4_t b = __builtin_nontemporal_load(tp + 1);
    vint4_t c = __builtin_nontemporal_load(tp + 2);
    unsigned full = ((unsigned)a.x << 11) | ((unsigned)a.y << 10) |
                    ((unsigned)a.z << 9)  | ((unsigned)a.w << 8)  |
                    ((unsigned)b.x << 7)  | ((unsigned)b.y << 6)  |
                    ((unsigned)b.z << 5)  | ((unsigned)b.w << 4)  |
                    ((unsigned)c.x << 3)  | ((unsigned)c.y << 2)  |
                    ((unsigned)c.z << 1)  | ((unsigned)c.w);
    vfloat4_t o0, o1, o2, o3, o4, o5, o6, o7;
    {  // s = 0 : in3 = 3
      unsigned paddr = full & 7u;
      unsigned p3 = pb_lds[0 + paddr];
      o0.x = (float)(p3 & 1u); o0.y = (float)((p3 >> 1) & 1u); o0.z = (float)((p3 >> 2) & 1u);
      unsigned w = paddr | (p3 << 3);
      o0.w = rt_lds[0 + w];
      o1.x = rt_lds[64 + w];  o1.y = rt_lds[128 + w];
      o1.z = rt_lds[192 + w]; o1.w = rt_lds[256 + w];
    }
    {  // s = 1 : in3 = 6
      unsigned paddr = full & 63u;
      unsigned p3 = pb_lds[8 + paddr];
      o2.x = (float)(p3 & 1u); o2.y = (float)((p3 >> 1) & 1u); o2.z = (float)((p3 >> 2) & 1u);
      unsigned w = paddr | (p3 << 6);
      o2.w = rt_lds[320 + w];
      o3.x = rt_lds[320 + 512 + w];  o3.y = rt_lds[320 + 1024 + w];
      o3.z = rt_lds[320 + 1536 + w]; o3.w = rt_lds[320 + 2048 + w];
    }
    {  // s = 2 : in3 = 9
      unsigned paddr = full & 511u;
      unsigned p3 = pb_lds[72 + paddr];
      o4.x = (float)(p3 & 1u); o4.y = (float)((p3 >> 1) & 1u); o4.z = (float)((p3 >> 2) & 1u);
      unsigned w = paddr | (p3 << 9);
      o4.w = rt_lds[2880 + w];
      o5.x = rt_lds[2880 + 4096 + w];  o5.y = rt_lds[2880 + 8192 + w];
      o5.z = rt_lds[2880 + 12288 + w]; o5.w = rt_lds[2880 + 16384 + w];
    }
    {  // s = 3 : in3 = 12, RT3 from global (L2-resident 640 KB)
      unsigned paddr = full & 4095u;
      unsigned p3 = pb_lds[584 + paddr];
      o6.x = (float)(p3 & 1u); o6.y = (float)((p3 >> 1) & 1u); o6.z = (float)((p3 >> 2) & 1u);
      unsigned w = paddr | (p3 << 12);
      o6.w = rt3[w];
      o7.x = rt3[32768 + w];  o7.y = rt3[65536 + w];
      o7.z = rt3[98304 + w];  o7.w = rt3[131072 + w];
    }
    vfloat4_t* op = (vfloat4_t*)(out + (size_t)i * 32);
    __builtin_nontemporal_store(o0, op + 0);
    __builtin_nontemporal_store(o1, op + 1);
    __builtin_nontemporal_store(o2, op + 2);
    __builtin_nontemporal_store(o3, op + 3);
    __builtin_nontemporal_store(o4, op + 4);
    __builtin_nontemporal_store(o5, op + 5);
    __builtin_nontemporal_store(o6, op + 6);
    __builtin_nontemporal_store(o7, op + 7);
  }
}

// ---------------------------------------------------------------------------
// Fallback (no precompute scratch): direct evaluation, conn%in_bits in LDS.
// ---------------------------------------------------------------------------
__global__ __launch_bounds__(256) void msp_direct_kernel(
    const int* __restrict__ tb, const float* __restrict__ pat,
    const float* __restrict__ pos, const int* __restrict__ conn,
    float* __restrict__ out, int nitems) {
  __shared__ unsigned char cmod[240];
  for (int i = threadIdx.x; i < 240; i += blockDim.x) {
    int s = i / 60;
    int in_bits = 3 * (s + 1) + 3;
    cmod[i] = (unsigned char)(conn[i] % in_bits);
  }
  __syncthreads();
  int stride = gridDim.x * blockDim.x;
  for (int i = blockIdx.x * blockDim.x + threadIdx.x; i < nitems; i += stride) {
    const vint4_t* tp = (const vint4_t*)(tb + (size_t)i * 12);
    vint4_t a = tp[0], b = tp[1], c = tp[2];
    unsigned full = ((unsigned)a.x << 11) | ((unsigned)a.y << 10) |
                    ((unsigned)a.z << 9)  | ((unsigned)a.w << 8)  |
                    ((unsigned)b.x << 7)  | ((unsigned)b.y << 6)  |
                    ((unsigned)b.z << 5)  | ((unsigned)b.w << 4)  |
                    ((unsigned)c.x << 3)  | ((unsigned)c.y << 2)  |
                    ((unsigned)c.z << 1)  | ((unsigned)c.w);
    float* op = out + (size_t)i * 32;
#pragma unroll
    for (int s = 0; s < 4; ++s) {
      int in3 = 3 * (s + 1);
      int in_bits = in3 + 3;
      int nb = in_bits < 12 ? in_bits : 12;
      unsigned paddr = full & ((1u << in3) - 1u);
      unsigned bits = 0;
#pragma unroll
      for (int t = 0; t < 3; ++t) {
        float v = pat[(s * 3 + t) * 4096 + paddr];
        bits |= (v > 0.5f ? 1u : 0u) << t;
        op[s * 8 + t] = (v > 0.5f) ? 1.0f : 0.0f;
      }
      unsigned w = paddr | (bits << in3);
#pragma unroll
      for (int k = 0; k < 5; ++k) {
        int addr = 0;
        for (int j = 0; j < nb; ++j) {
          int cc = cmod[s * 60 + k * 12 + j];
          int bit = (cc < in3) ? ((w >> (in3 - 1 - cc)) & 1) : ((w >> cc) & 1);
          addr += bit << j;
        }
        op[s * 8 + 3 + k] = pos[(s * 5 + k) * 4096 + addr];
      }
    }
  }
}

extern "C" void kernel_launch(void* const* d_in, const int* in_sizes, int n_in,
                              void* d_out, int out_size, void* d_ws, size_t ws_size,
                              hipStream_t stream) {
  const int*   tb   = (const int*)d_in[0];
  const float* pat  = (const float*)d_in[1];
  const float* pos  = (const float*)d_in[2];
  const int*   conn = (const int*)d_in[3];
  float* out = (float*)d_out;
  int nitems = in_sizes[0] / 12;  // B

  size_t need = (size_t)RT_TOTAL * 4 + PB_TOTAL;
  if (ws_size >= need) {
    float* rt = (float*)d_ws;
    unsigned char* pbg = (unsigned char*)d_ws + (size_t)RT_TOTAL * 4;
    build_pb_kernel<<<(PB_TOTAL + 255) / 256, 256, 0, stream>>>(pat, pbg);
    build_rt_kernel<<<(RT_TOTAL + 255) / 256, 256, 0, stream>>>(pos, conn, rt);
    msp_main_kernel<<<512, 1024, 0, stream>>>(tb, rt, pbg, out, nitems);
  } else {
    msp_direct_kernel<<<1024, 256, 0, stream>>>(tb, pat, pos, conn, out, nitems);
  }
}